// ClebschCombiningSingleUnrolled_60163901882614
// MI455X (gfx1250) — compile-verified
//
#include <hip/hip_runtime.h>
#include <hip/hip_bf16.h>

// Clebsch-Gordan combine: out[m,c] = sum_{m1+m2=m} cg[m1,m2]*X1[m1,c]*X2[m2,c]
// M = L = 7 (lambd = 3), c ranges over N*F flattened columns.
// Pure streaming kernel (~1 flop/byte): async global->LDS fills (ASYNCcnt),
// NT temporal hints both directions, b128 everywhere.

#define MDIM 7
#define TPB  256
#define VEC  4
#define TILE (TPB * VEC)   // 1024 columns per block

typedef float v4f __attribute__((ext_vector_type(4)));

__global__ __launch_bounds__(TPB) void cg_combine_async_kernel(
    const float* __restrict__ X1,
    const float* __restrict__ X2,
    const float* __restrict__ CG,
    float* __restrict__ OUT,
    int nf)
{
    __shared__ __align__(16) float sx1[MDIM][TILE];
    __shared__ __align__(16) float sx2[MDIM][TILE];

    const int tid = threadIdx.x;
    const int c0  = blockIdx.x * TILE + tid * VEC;

    if (c0 + VEC <= nf) {
        // LDS byte addresses of this thread's 16B slots (low 32 bits of the
        // generic address of a __shared__ object == LDS byte offset).
        const unsigned lds1 = (unsigned)(unsigned long long)(uintptr_t)&sx1[0][tid * VEC];
        const unsigned lds2 = (unsigned)(unsigned long long)(uintptr_t)&sx2[0][tid * VEC];
        const unsigned voff = (unsigned)c0 * 4u;
        const unsigned rowb = (unsigned)nf * 4u;   // global row stride in bytes

        // 14 async b128 fills per thread: X1[m][c0..c0+3], X2[m][c0..c0+3].
        // Non-temporal: 352 MB stream with zero reuse, keep it out of L2.
        #pragma unroll
        for (int m = 0; m < MDIM; ++m) {
            const unsigned g  = voff + (unsigned)m * rowb;
            const unsigned l1 = lds1 + (unsigned)(m * (TILE * 4));
            const unsigned l2 = lds2 + (unsigned)(m * (TILE * 4));
            asm volatile("global_load_async_to_lds_b128 %0, %1, %2 th:TH_LOAD_NT"
                         :: "v"(l1), "v"(g), "s"(X1) : "memory");
            asm volatile("global_load_async_to_lds_b128 %0, %1, %2 th:TH_LOAD_NT"
                         :: "v"(l2), "v"(g), "s"(X2) : "memory");
        }

        // Wait for this wave's async fills; each lane reads back only the LDS
        // slots it filled itself, so no workgroup barrier is required.
        asm volatile("s_wait_asynccnt 0" ::: "memory");

        v4f acc[MDIM];
        #pragma unroll
        for (int m = 0; m < MDIM; ++m)
            acc[m] = (v4f){0.f, 0.f, 0.f, 0.f};

        #pragma unroll
        for (int m1 = 0; m1 < MDIM; ++m1) {
            const v4f a = *reinterpret_cast<const v4f*>(&sx1[m1][tid * VEC]);
            #pragma unroll
            for (int m2 = 0; m1 + m2 < MDIM; ++m2) {
                const float c = CG[m1 * MDIM + m2];
                const v4f   b = *reinterpret_cast<const v4f*>(&sx2[m2][tid * VEC]);
                const int  mo = m1 + m2;
                acc[mo].x = fmaf(c * a.x, b.x, acc[mo].x);
                acc[mo].y = fmaf(c * a.y, b.y, acc[mo].y);
                acc[mo].z = fmaf(c * a.z, b.z, acc[mo].z);
                acc[mo].w = fmaf(c * a.w, b.w, acc[mo].w);
            }
        }

        #pragma unroll
        for (int m = 0; m < MDIM; ++m) {
            v4f* dst = reinterpret_cast<v4f*>(OUT + (size_t)m * (size_t)nf + c0);
            __builtin_nontemporal_store(acc[m], dst);
        }
    } else if (c0 < nf) {
        // Scalar tail (nf not a multiple of VEC) — direct global path.
        for (int c = c0; c < c0 + VEC && c < nf; ++c) {
            float o[MDIM];
            #pragma unroll
            for (int m = 0; m < MDIM; ++m) o[m] = 0.f;
            #pragma unroll
            for (int m1 = 0; m1 < MDIM; ++m1) {
                const float a = X1[(size_t)m1 * (size_t)nf + c];
                #pragma unroll
                for (int m2 = 0; m1 + m2 < MDIM; ++m2) {
                    o[m1 + m2] = fmaf(CG[m1 * MDIM + m2] * a,
                                      X2[(size_t)m2 * (size_t)nf + c],
                                      o[m1 + m2]);
                }
            }
            #pragma unroll
            for (int m = 0; m < MDIM; ++m)
                OUT[(size_t)m * (size_t)nf + c] = o[m];
        }
    }
}

extern "C" void kernel_launch(void* const* d_in, const int* in_sizes, int n_in,
                              void* d_out, int out_size, void* d_ws, size_t ws_size,
                              hipStream_t stream) {
    const float* X1 = (const float*)d_in[0];   // [7, N, F] f32
    const float* X2 = (const float*)d_in[1];   // [7, N, F] f32
    const float* CG = (const float*)d_in[2];   // [7, 7]    f32
    // d_in[3] = lambd (==3 -> L = 7 = MDIM), baked into the unrolled kernel.
    float* OUT = (float*)d_out;                // [7, N, F] f32

    const int nf = in_sizes[0] / MDIM;         // N*F flattened columns
    const int blocks = (nf + TILE - 1) / TILE;

    hipLaunchKernelGGL(cg_combine_async_kernel,
                       dim3(blocks), dim3(TPB), 0, stream,
                       X1, X2, CG, OUT, nf);
}